// Net_21474836480123
// MI455X (gfx1250) — compile-verified
//
#include <hip/hip_runtime.h>
#include <math.h>

typedef __attribute__((ext_vector_type(2))) float v2f;
typedef __attribute__((ext_vector_type(8))) float v8f;

#define N_NODES 100000
#define N_EDGES 1600000
#define F_IN    128
#define HID     64
#define N_CLS   40
#define LD3     48   // layer-3 padded leading dim (40 -> 48 = 3 WMMA N-tiles)

// ---------------- degree / normalization ----------------

__global__ void k_init_deg(float* __restrict__ deg) {
  int i = blockIdx.x * blockDim.x + threadIdx.x;
  if (i < N_NODES) deg[i] = 1.0f;  // self-loop contributes 1 to every node
}

__global__ void k_deg(const int* __restrict__ dst, float* __restrict__ deg) {
  int e = blockIdx.x * blockDim.x + threadIdx.x;
  if (e < N_EDGES) atomicAdd(&deg[dst[e]], 1.0f);  // global_atomic_add_f32 (no return)
}

__global__ void k_dinv(const float* __restrict__ deg, float* __restrict__ dinv) {
  int i = blockIdx.x * blockDim.x + threadIdx.x;
  if (i < N_NODES) dinv[i] = rsqrtf(deg[i]);  // deg >= 1 always
}

// ---------------- W3 zero-pad: [64,40] -> [64,48] ----------------
__global__ void k_pad_w3(const float* __restrict__ W3, float* __restrict__ W3p) {
  int idx = blockIdx.x * blockDim.x + threadIdx.x;   // over 64*48
  if (idx >= HID * LD3) return;
  int r = idx / LD3, c = idx - r * LD3;
  W3p[idx] = (c < N_CLS) ? W3[r * N_CLS + c] : 0.0f;
}

// ---------------- fp32 WMMA GEMM: C[M,N] = A[M,K] @ B[K,N] ----------------
// Block = 128 threads (4 waves), gridDim.x = M/16 (100000 = 6250*16 exact).
// Wave w computes the 16x16 tile at columns [16w,16w+16); extra waves exit whole-wave
// (EXEC all-ones for WMMA). Compile-time strides -> immediate-offset global loads,
// no guards, full K unroll (K/4 x V_WMMA_F32_16X16X4_F32).
// A operand (16x4 f32, 2 VGPRs): lanes 0-15 -> K pair {0,1}; lanes 16-31 -> {2,3}.
// B operand (4x16 f32, 2 VGPRs): same K split, N = lane&15.
template <int K, int LDA, int LDB, int LDC, bool RELU>
__global__ __launch_bounds__(128)
void k_gemm(const float* __restrict__ A, const float* __restrict__ B,
            float* __restrict__ C, int nTilesN) {
  const int wave = threadIdx.x >> 5;
  if (wave >= nTilesN) return;                 // uniform per wave
  const int lane  = threadIdx.x & 31;
  const int half  = lane >> 4;
  const int l     = lane & 15;
  const int m0    = blockIdx.x * 16;
  const int tileN = wave * 16;

  const float* ap = A + (m0 + l) * LDA + 2 * half;   // + k       -> A[m0+l][k+2*half]
  const float* bp = B + 2 * half * LDB + tileN + l;  // + k*LDB   -> B[k+2*half][tileN+l]

  v8f acc = {};
#pragma unroll
  for (int k = 0; k < K; k += 4) {
    float a0 = ap[k], a1 = ap[k + 1];
    if (RELU) { a0 = fmaxf(a0, 0.0f); a1 = fmaxf(a1, 0.0f); }
    v2f av = {a0, a1};
    v2f bv = {bp[k * LDB], bp[(k + 1) * LDB]};
    // (neg_a, A, neg_b, B, c_mod, C, reuse_a, reuse_b)
    acc = __builtin_amdgcn_wmma_f32_16x16x4_f32(false, av, false, bv,
                                                (short)0, acc, false, false);
  }
  // C/D layout: VGPR r -> row m0 + 8*half + r, col tileN + (lane&15)
  float* cp = C + (m0 + 8 * half) * LDC + tileN + l;
#pragma unroll
  for (int r = 0; r < 8; ++r) cp[r * LDC] = acc[r];
}

// ---------------- self-loop message + bias (no atomics) ----------------
// out[i,f] = h[i,f] * dinv[i]^2 + bias[f]   (F cols written, LD row stride)
template <int F, int LD>
__global__ void k_selfloop_bias(const float* __restrict__ h,
                                const float* __restrict__ dinv,
                                const float* __restrict__ bias,
                                float* __restrict__ out) {
  int idx = blockIdx.x * blockDim.x + threadIdx.x;   // over N_NODES*F (< 2^31)
  if (idx >= N_NODES * F) return;
  int i = idx / F;
  int f = idx - i * F;
  float di = dinv[i];
  out[i * LD + f] = h[i * LD + f] * di * di + bias[f];
}

// ---------------- edge scatter: out[dst] += h[src] * dinv[src]*dinv[dst] ----------
// FCH threads per edge, each does one float4 gather + 4 fp32 global atomics.
// Both feature matrices are L2-resident (25.6 MB << 192 MB L2).
template <int FCH, int LD>
__global__ void k_scatter(const int* __restrict__ src, const int* __restrict__ dst,
                          const float* __restrict__ dinv,
                          const float* __restrict__ h, float* __restrict__ out) {
  int t = blockIdx.x * blockDim.x + threadIdx.x;
  int e = t / FCH;
  if (e >= N_EDGES) return;
  int c = t - e * FCH;
  int s = src[e], d = dst[e];
  float norm = dinv[s] * dinv[d];
  const float4 hv = *(const float4*)(h + s * LD + c * 4);
  float* o = out + d * LD + c * 4;
  atomicAdd(o + 0, hv.x * norm);
  atomicAdd(o + 1, hv.y * norm);
  atomicAdd(o + 2, hv.z * norm);
  atomicAdd(o + 3, hv.w * norm);
}

// ---------------- fused ReLU + log_softmax over 40 classes ----------------
__global__ void k_relu_logsoftmax(const float* __restrict__ a, float* __restrict__ out) {
  int i = blockIdx.x * blockDim.x + threadIdx.x;
  if (i >= N_NODES) return;
  const float* row = a + i * LD3;
  float v[N_CLS];
  float m = 0.0f;  // relu'd values are >= 0
#pragma unroll
  for (int c = 0; c < N_CLS; ++c) { v[c] = fmaxf(row[c], 0.0f); m = fmaxf(m, v[c]); }
  float s = 0.0f;
#pragma unroll
  for (int c = 0; c < N_CLS; ++c) s += expf(v[c] - m);
  float ls = logf(s) + m;
  float* orow = out + i * N_CLS;
#pragma unroll
  for (int c = 0; c < N_CLS; ++c) orow[c] = v[c] - ls;
}

// ---------------- launch ----------------

extern "C" void kernel_launch(void* const* d_in, const int* in_sizes, int n_in,
                              void* d_out, int out_size, void* d_ws, size_t ws_size,
                              hipStream_t stream) {
  (void)in_sizes; (void)n_in; (void)out_size; (void)ws_size;
  const float* x   = (const float*)d_in[0];
  const int*   ei  = (const int*)d_in[1];          // [2, E]: row 0 = src, row 1 = dst
  const int*   src = ei;
  const int*   dst = ei + N_EDGES;
  const float* W1  = (const float*)d_in[2];
  const float* b1  = (const float*)d_in[3];
  const float* W2  = (const float*)d_in[4];
  const float* b2  = (const float*)d_in[5];
  const float* W3  = (const float*)d_in[6];
  const float* b3  = (const float*)d_in[7];
  float* out = (float*)d_out;

  // workspace layout (each region starts 16B aligned)
  float* deg  = (float*)d_ws;                      // [N]
  float* dinv = deg  + N_NODES;                    // [N]
  float* bufH = dinv + N_NODES;                    // [N,64] GEMM out (layer3: stride 48)
  float* bufA = bufH + (size_t)N_NODES * HID;      // [N,64] aggregate (layer3: stride 48)
  float* W3p  = bufA + (size_t)N_NODES * HID;      // [64,48] zero-padded W3

  const int TB = 256;
  // degrees + normalization + W3 padding
  k_init_deg<<<(N_NODES + TB - 1) / TB, TB, 0, stream>>>(deg);
  k_deg<<<(N_EDGES + TB - 1) / TB, TB, 0, stream>>>(dst, deg);
  k_dinv<<<(N_NODES + TB - 1) / TB, TB, 0, stream>>>(deg, dinv);
  k_pad_w3<<<(HID * LD3 + TB - 1) / TB, TB, 0, stream>>>(W3, W3p);

  const dim3 gemmGrid(N_NODES / 16);  // 6250, exact
  const int nf64 = N_NODES * HID;
  const int nf40 = N_NODES * N_CLS;

  // ---- layer 1: h = x @ W1 ; agg = selfloop + b1 + scatter(h) ----
  k_gemm<F_IN, F_IN, HID, HID, false><<<gemmGrid, 128, 0, stream>>>(x, W1, bufH, HID / 16);
  k_selfloop_bias<HID, HID><<<(nf64 + TB - 1) / TB, TB, 0, stream>>>(bufH, dinv, b1, bufA);
  k_scatter<HID / 4, HID><<<(N_EDGES * (HID / 4) + TB - 1) / TB, TB, 0, stream>>>(src, dst, dinv, bufH, bufA);

  // ---- layer 2: h = relu(agg) @ W2 (ReLU fused into A loads) ----
  k_gemm<HID, HID, HID, HID, true><<<gemmGrid, 128, 0, stream>>>(bufA, W2, bufH, HID / 16);
  k_selfloop_bias<HID, HID><<<(nf64 + TB - 1) / TB, TB, 0, stream>>>(bufH, dinv, b2, bufA);
  k_scatter<HID / 4, HID><<<(N_EDGES * (HID / 4) + TB - 1) / TB, TB, 0, stream>>>(src, dst, dinv, bufH, bufA);

  // ---- layer 3: h = relu(agg) @ W3p (N padded 40 -> 48, no guards) ----
  k_gemm<HID, HID, LD3, LD3, true><<<gemmGrid, 128, 0, stream>>>(bufA, W3p, bufH, LD3 / 16);
  k_selfloop_bias<N_CLS, LD3><<<(nf40 + TB - 1) / TB, TB, 0, stream>>>(bufH, dinv, b3, bufA);
  k_scatter<N_CLS / 4, LD3><<<(N_EDGES * (N_CLS / 4) + TB - 1) / TB, TB, 0, stream>>>(src, dst, dinv, bufH, bufA);

  // ---- relu + log_softmax -> d_out [N, 40] ----
  k_relu_logsoftmax<<<(N_NODES + TB - 1) / TB, TB, 0, stream>>>(bufA, out);
}